// ExplorerV1_9294309229175
// MI455X (gfx1250) — compile-verified
//
#include <hip/hip_runtime.h>
#include <hip/hip_bf16.h>

// Mamba-bidir, tiny dims: one workgroup per batch row, whole sequence in LDS,
// WMMA f32 16x16x4 for position-parallel projections, wave0 runs the scan,
// TDM tensor_load_to_lds for the initial sequence staging (when available).

#define NLAYERS 4
#define DMODEL  2
#define DINNER  4
#define DSTATE  16
#define DCONV   4
#define LSEQ    1024
#define CH      128            // chunk of positions processed per phase
#define NCHUNK  (LSEQ / CH)
#define EPSR    1e-5f

typedef float v2f  __attribute__((ext_vector_type(2)));
typedef float v8f  __attribute__((ext_vector_type(8)));
typedef unsigned int u32x4 __attribute__((ext_vector_type(4)));
typedef int   i32x4 __attribute__((ext_vector_type(4)));
typedef int   i32x8 __attribute__((ext_vector_type(8)));

#if defined(__gfx1250__) && __has_builtin(__builtin_amdgcn_tensor_load_to_lds) && \
    __has_builtin(__builtin_amdgcn_s_wait_tensorcnt)
#define HAVE_TDM 1
#else
#define HAVE_TDM 0
#endif

// Native-rate transcendentals (TRANS ops co-execute with VALU on CDNA5).
__device__ __forceinline__ float rcp_fast(float x) {
#if __has_builtin(__builtin_amdgcn_rcpf)
    return __builtin_amdgcn_rcpf(x);
#else
    return 1.0f / x;
#endif
}
__device__ __forceinline__ float rsq_fast(float x) {
#if __has_builtin(__builtin_amdgcn_rsqf)
    return __builtin_amdgcn_rsqf(x);
#else
    return rsqrtf(x);
#endif
}
__device__ __forceinline__ float sigmoidf_(float x) {
    return rcp_fast(1.0f + __expf(-x));
}
__device__ __forceinline__ float softplusf_(float x) {
    // log(1+e^x); for x>20 the correction underflows fp32.
    return (x > 20.0f) ? x : __logf(1.0f + __expf(x));
}

#if HAVE_TDM
// Issue a TDM copy of `n` contiguous 4-byte elements from `gsrc` to LDS offset
// `lds_off` (bytes). D# per CDNA5 ISA §8: group0 {count=1, lds_addr,
// global_addr, type=2}; group1 {data_size=4B, tensor nx1, tile nx1}.
// This toolchain exposes the 6-arg builtin form:
//   (u32x4 g0, i32x8 g1, i32x4 g2, i32x4 g3, i32x8 extra, i32 cpol)
// groups 2/3/extra are zero (NULL) for a <=2D tensor.
__device__ __forceinline__ void tdm_load_row(const float* gsrc, unsigned lds_off, int n) {
    unsigned long long ga = (unsigned long long)(const void*)gsrc;
    u32x4 g0;
    g0[0] = 1u;                                   // count=1, no gather/restore
    g0[1] = lds_off;                              // LDS byte address
    g0[2] = (unsigned)(ga & 0xFFFFFFFFu);         // global_addr[31:0]
    g0[3] = (unsigned)((ga >> 32) & 0x1FFFFFFu)   // global_addr[56:32]
          | (2u << 30);                           // type=2 (image)
    i32x8 g1;
    g1[0] = (2 << 16);                            // data_size=2 -> 4 bytes
    g1[1] = (n & 0xFFFF) << 16;                   // tensor_dim0[15:0]
    g1[2] = ((n >> 16) & 0xFFFF) | (1 << 16);     // tensor_dim0[31:16], tensor_dim1=1
    g1[3] = (n & 0xFFFF) << 16;                   // tile_dim0 = n
    g1[4] = 1;                                    // tile_dim1=1, tile_dim2 unused
    g1[5] = n;                                    // tensor_dim0_stride[31:0]
    g1[6] = 0;                                    // stride hi / dim1_stride
    g1[7] = 0;
    i32x4 z4 = {0, 0, 0, 0};
    i32x8 z8 = {0, 0, 0, 0, 0, 0, 0, 0};
    __builtin_amdgcn_tensor_load_to_lds(g0, g1, z4, z4, z8, 0);
}
#endif

__global__ __launch_bounds__(CH) void mamba_bidir_kernel(
    const float* __restrict__ intensity,   // (B, L)
    const float* __restrict__ angle,       // (B, L)
    const float* __restrict__ in_proj_w,   // (4,2, 8,2)
    const float* __restrict__ conv_w,      // (4,2, 4,4)
    const float* __restrict__ conv_b,      // (4,2, 4)
    const float* __restrict__ x_proj_w,    // (4,2, 33,4)
    const float* __restrict__ dt_proj_w,   // (4,2, 4,1)
    const float* __restrict__ dt_proj_b,   // (4,2, 4)
    const float* __restrict__ A_log,       // (4,2, 4,16)
    const float* __restrict__ D_param,     // (4,2, 4)
    const float* __restrict__ out_proj_w,  // (4,2, 2,4)
    const float* __restrict__ norm_w,      // (4, 2)
    const float* __restrict__ norm_f_w,    // (2,)
    float* __restrict__ out)               // (B, L, 2)
{
    const int b    = blockIdx.x;
    const int tid  = threadIdx.x;
    const int lane = tid & 31;
    const int wave = tid >> 5;

    __shared__ float s_x0 [LSEQ];             // residual stream (SoA)
    __shared__ float s_x1 [LSEQ];
    __shared__ float s_xn0[LSEQ];             // rmsnorm'd input
    __shared__ float s_xn1[LSEQ];
    __shared__ float s_xz [CH + 3][16];       // in-proj tile: cols 0-3 xi-preconv
                                              // (rows shifted +3 = conv halo),
                                              // cols 4-7 gate z, cols 8-15 zero
    __shared__ float s_xi [CH][DINNER];       // post conv+silu
    __shared__ float s_dl [CH][DINNER];       // delta (softplus)
    __shared__ float s_B  [CH][DSTATE];
    __shared__ float s_C  [CH][DSTATE];
    __shared__ float s_y  [CH][DINNER];

    // ---- stage sequence into LDS ----
#if HAVE_TDM
    if (wave == 0) {
        tdm_load_row(intensity + (size_t)b * LSEQ, (unsigned)(size_t)(void*)s_x0, LSEQ);
        tdm_load_row(angle     + (size_t)b * LSEQ, (unsigned)(size_t)(void*)s_x1, LSEQ);
        __builtin_amdgcn_s_wait_tensorcnt(0);
    }
#else
    for (int p = tid; p < LSEQ; p += CH) {
        s_x0[p] = intensity[(size_t)b * LSEQ + p];
        s_x1[p] = angle    [(size_t)b * LSEQ + p];
    }
#endif
    __syncthreads();

    for (int layer = 0; layer < NLAYERS; ++layer) {
        // ---- rmsnorm over D_MODEL=2 ----
        const float nw0 = norm_w[layer * 2 + 0];
        const float nw1 = norm_w[layer * 2 + 1];
        for (int p = tid; p < LSEQ; p += CH) {
            float a = s_x0[p], c = s_x1[p];
            float r = rsq_fast(0.5f * (a * a + c * c) + EPSR);
            s_xn0[p] = a * r * nw0;
            s_xn1[p] = c * r * nw1;
        }
        __syncthreads();

        for (int dir = 0; dir < 2; ++dir) {
            const int ld = layer * 2 + dir;
            const float* Win  = in_proj_w  + ld * (2 * DINNER) * DMODEL; // (8,2)
            const float* Cw   = conv_w     + ld * DINNER * DCONV;        // (4,4)
            const float* Cb   = conv_b     + ld * DINNER;
            const float* Xp   = x_proj_w   + ld * (1 + 2 * DSTATE) * DINNER; // (33,4)
            const float* Dtw  = dt_proj_w  + ld * DINNER;                // (4,1)
            const float* Dtb  = dt_proj_b  + ld * DINNER;
            const float* Al   = A_log      + ld * DINNER * DSTATE;       // (4,16)
            const float* Dp   = D_param    + ld * DINNER;
            const float* Wout = out_proj_w + ld * DMODEL * DINNER;       // (2,4)

            // scan-lane mapping: lane -> (n = lane&15, d-pair = (lane>>4)*2)
            const int   n     = lane & 15;
            const int   dpair = (lane >> 4) * 2;
            const float An0   = -__expf(Al[(dpair + 0) * DSTATE + n]);
            const float An1   = -__expf(Al[(dpair + 1) * DSTATE + n]);
            float h0 = 0.0f, h1 = 0.0f;

            // WMMA B operands (constant across chunks), branchless selects.
            // B 4x16 (KxN): v0 holds K=0 (lanes 0-15) / K=2 (lanes 16-31); v1: K=1/K=3.
            const int  nn   = lane & 15;
            const int  k0   = (lane >> 4) * 2;      // 0 or 2
            const bool kLow = (lane < 16);
            v2f b_in, b_B, b_C;
            // in_proj: B[k][j] = Win[j][k]; rows k>=2 and cols j>=8 are zero pad.
            b_in.x = (kLow && nn < 2 * DINNER) ? Win[nn * DMODEL + 0] : 0.0f;
            b_in.y = (kLow && nn < 2 * DINNER) ? Win[nn * DMODEL + 1] : 0.0f;
            // B/C proj: all 4 K rows real.
            b_B.x = Xp[(1 + nn) * DINNER + k0];
            b_B.y = Xp[(1 + nn) * DINNER + k0 + 1];
            b_C.x = Xp[(1 + DSTATE + nn) * DINNER + k0];
            b_C.y = Xp[(1 + DSTATE + nn) * DINNER + k0 + 1];

            const int mrow = lane & 15;             // WMMA A: M = lane&15
            const int rofs = (lane >> 4) << 3;      // D rows: +0 / +8 per half

            for (int chunk = 0; chunk < NCHUNK; ++chunk) {
                const int start = chunk * CH;

                // ---- A1a: conv halo (3 positions before chunk, scalar) ----
                if (tid < 3) {
                    int tpos = start - 3 + tid;
                    float v0 = 0.f, v1 = 0.f, v2 = 0.f, v3 = 0.f;
                    if (tpos >= 0) {
                        int ip = dir ? (LSEQ - 1 - tpos) : tpos;
                        float a = s_xn0[ip], c = s_xn1[ip];
                        v0 = a * Win[0] + c * Win[1];
                        v1 = a * Win[2] + c * Win[3];
                        v2 = a * Win[4] + c * Win[5];
                        v3 = a * Win[6] + c * Win[7];
                    }
                    s_xz[tid][0] = v0; s_xz[tid][1] = v1;
                    s_xz[tid][2] = v2; s_xz[tid][3] = v3;
                }

                // ---- A1b: in-projection via WMMA f32 16x16x4 (2 tiles/wave) ----
                for (int tt = wave; tt < CH / 16; tt += 4) {
                    int t_scan = start + tt * 16 + mrow;
                    int ip     = dir ? (LSEQ - 1 - t_scan) : t_scan;
                    // K=2,3 B-rows are zero, so lanes 16-31 may carry anything.
                    v2f a; a.x = s_xn0[ip]; a.y = s_xn1[ip];
                    v8f c = {};
                    c = __builtin_amdgcn_wmma_f32_16x16x4_f32(
                            false, a, false, b_in, (short)0, c, false, false);
                    // Unconditional 16-wide scatter: rows r -> positions
                    // tt*16 + r + rofs (shift +3 for conv halo), all 16 cols.
                    #pragma unroll
                    for (int r = 0; r < 8; ++r)
                        s_xz[tt * 16 + r + rofs + 3][nn] = c[r];
                }
                __syncthreads();

                // ---- A2: causal conv + silu + delta (thread == position) ----
                {
                    int p = tid;
                    float xiv[DINNER];
                    #pragma unroll
                    for (int d = 0; d < DINNER; ++d) {
                        float acc = Cb[d];
                        #pragma unroll
                        for (int k = 0; k < DCONV; ++k)
                            acc += Cw[d * DCONV + k] * s_xz[p + k][d];
                        xiv[d] = acc * sigmoidf_(acc);
                        s_xi[p][d] = xiv[d];
                    }
                    float dt = 0.0f;
                    #pragma unroll
                    for (int d = 0; d < DINNER; ++d) dt += xiv[d] * Xp[d]; // x_proj row 0
                    #pragma unroll
                    for (int di = 0; di < DINNER; ++di)
                        s_dl[p][di] = softplusf_(dt * Dtw[di] + Dtb[di]);
                }
                __syncthreads();

                // ---- A3: B/C projections via WMMA (branchless operands) ----
                for (int tt = wave; tt < CH / 16; tt += 4) {
                    int p = tt * 16 + mrow;
                    v2f a; a.x = s_xi[p][k0]; a.y = s_xi[p][k0 + 1]; // K rows 0,1 / 2,3
                    v8f cB = {}, cC = {};
                    cB = __builtin_amdgcn_wmma_f32_16x16x4_f32(
                            false, a, false, b_B, (short)0, cB, false, false);
                    cC = __builtin_amdgcn_wmma_f32_16x16x4_f32(
                            false, a, false, b_C, (short)0, cC, false, false);
                    #pragma unroll
                    for (int r = 0; r < 8; ++r) {
                        int p2 = tt * 16 + r + rofs;
                        s_B[p2][nn] = cB[r];
                        s_C[p2][nn] = cC[r];
                    }
                }
                __syncthreads();

                // ---- B: sequential scan (wave 0; 2 states per lane) ----
                if (wave == 0) {
                    #pragma unroll 4
                    for (int p = 0; p < CH; ++p) {
                        float d0  = s_dl[p][dpair], d1 = s_dl[p][dpair + 1];
                        float xv0 = s_xi[p][dpair], xv1 = s_xi[p][dpair + 1];
                        float Bn  = s_B[p][n];
                        float Cn  = s_C[p][n];
                        h0 = __expf(d0 * An0) * h0 + d0 * Bn * xv0;
                        h1 = __expf(d1 * An1) * h1 + d1 * Bn * xv1;
                        float y0 = h0 * Cn, y1 = h1 * Cn;
                        // reduce over the 16 n-lanes (stays within each half-wave)
                        #pragma unroll
                        for (int msk = 1; msk < 16; msk <<= 1) {
                            y0 += __shfl_xor(y0, msk, 32);
                            y1 += __shfl_xor(y1, msk, 32);
                        }
                        if ((lane & 15) == 0) {
                            s_y[p][dpair]     = y0;
                            s_y[p][dpair + 1] = y1;
                        }
                    }
                }
                __syncthreads();

                // ---- C: skip/gate + out-projection + residual add ----
                {
                    int p      = tid;
                    int t_scan = start + p;
                    int ip     = dir ? (LSEQ - 1 - t_scan) : t_scan;
                    float o0 = 0.0f, o1 = 0.0f;
                    #pragma unroll
                    for (int d = 0; d < DINNER; ++d) {
                        float yv = s_y[p][d] + Dp[d] * s_xi[p][d];
                        float zz = s_xz[p + 3][DINNER + d];    // gate z
                        yv *= zz * sigmoidf_(zz);
                        o0 += yv * Wout[0 * DINNER + d];
                        o1 += yv * Wout[1 * DINNER + d];
                    }
                    s_x0[ip] += o0;
                    s_x1[ip] += o1;
                }
                __syncthreads();
            } // chunk
        } // dir
    } // layer

    // ---- final rmsnorm + vectorized store ----
    const float f0 = norm_f_w[0], f1 = norm_f_w[1];
    float2* out2 = (float2*)out;
    for (int p = tid; p < LSEQ; p += CH) {
        float a = s_x0[p], c = s_x1[p];
        float r = rsq_fast(0.5f * (a * a + c * c) + EPSR);
        out2[(size_t)b * LSEQ + p] = make_float2(a * r * f0, c * r * f1);
    }
}

extern "C" void kernel_launch(void* const* d_in, const int* in_sizes, int n_in,
                              void* d_out, int out_size, void* d_ws, size_t ws_size,
                              hipStream_t stream) {
    (void)n_in; (void)d_ws; (void)ws_size; (void)out_size;
    const float* intensity  = (const float*)d_in[0];
    const float* angle      = (const float*)d_in[1];
    const float* in_proj_w  = (const float*)d_in[2];
    const float* conv_w     = (const float*)d_in[3];
    const float* conv_b     = (const float*)d_in[4];
    const float* x_proj_w   = (const float*)d_in[5];
    const float* dt_proj_w  = (const float*)d_in[6];
    const float* dt_proj_b  = (const float*)d_in[7];
    const float* A_log      = (const float*)d_in[8];
    const float* D_param    = (const float*)d_in[9];
    const float* out_proj_w = (const float*)d_in[10];
    const float* norm_w     = (const float*)d_in[11];
    const float* norm_f_w   = (const float*)d_in[12];
    float* out = (float*)d_out;

    const int Bsz = in_sizes[0] / LSEQ;   // 512
    mamba_bidir_kernel<<<dim3(Bsz), dim3(CH), 0, stream>>>(
        intensity, angle, in_proj_w, conv_w, conv_b, x_proj_w,
        dt_proj_w, dt_proj_b, A_log, D_param, out_proj_w,
        norm_w, norm_f_w, out);
}